// NodeModel_84756884620023
// MI455X (gfx1250) — compile-verified
//
#include <hip/hip_runtime.h>
#include <math.h>

typedef float v2f __attribute__((ext_vector_type(2)));
typedef float v8f __attribute__((ext_vector_type(8)));

#define DH 128
#define DOUT 40
#define BN_EPS 1e-5f

// ---------------- degree / dinv ----------------
__global__ __launch_bounds__(256) void k_init_deg(float* __restrict__ deg, int n) {
  int i = blockIdx.x * 256 + threadIdx.x;
  if (i < n) deg[i] = 1.0f;
}

__global__ __launch_bounds__(256) void k_deg_accum(const int* __restrict__ dst,
                                                   float* __restrict__ deg, int e) {
  int i = blockIdx.x * 256 + threadIdx.x;
  if (i < e) atomicAdd(&deg[dst[i]], 1.0f);
}

__global__ __launch_bounds__(256) void k_dinv(float* __restrict__ deg, int n) {
  int i = blockIdx.x * 256 + threadIdx.x;
  if (i < n) deg[i] = rsqrtf(deg[i]);
}

// ---------------- weight transpose (W[k][n] -> Wt[n][k]) ----------------
__global__ __launch_bounds__(128) void k_transpose128(const float* __restrict__ W,
                                                      float* __restrict__ Wt) {
  int r = blockIdx.x, c = threadIdx.x;
  Wt[c * DH + r] = W[r * DH + c];
}

// ---------------- fp32 WMMA GEMM: C[nrows x 128] = A[nrows x 128] @ W ----
// Wt is W transposed (Wt[n][k] = W[k][n]) so B fragments are contiguous
// float2 along K.  One wave computes a 16-row stripe x 128 cols as eight
// 16x16 tiles, stepping K by 4 per v_wmma_f32_16x16x4_f32.
__global__ __launch_bounds__(256) void k_gemm128_wmma(const float* __restrict__ A,
                                                      const float* __restrict__ Wt,
                                                      float* __restrict__ C, int nrows) {
  const int lane = threadIdx.x & 31;
  const int wave = threadIdx.x >> 5;
  const int lm = lane & 15;   // A: row-in-tile, B: col, C/D: col
  const int kh = lane >> 4;   // k-half selector (K = kh*2 + j), C/D: row-half
  const int r0 = blockIdx.x * 128 + wave * 16;

  int arow = r0 + lm;
  if (arow >= nrows) arow = nrows - 1;            // clamp loads, mask stores
  const float* ap = A + (size_t)arow * DH + kh * 2;

  v8f acc[8] = {};                                 // zero accumulators
  for (int kb = 0; kb < 32; ++kb) {
    v2f a = *(const v2f*)(ap + kb * 4);            // A[m][kb*4 + kh*2 + {0,1}]
#pragma unroll
    for (int t = 0; t < 8; ++t) {
      v2f b = *(const v2f*)(Wt + (size_t)(t * 16 + lm) * DH + kb * 4 + kh * 2);
      acc[t] = __builtin_amdgcn_wmma_f32_16x16x4_f32(
          /*neg_a=*/false, a, /*neg_b=*/false, b,
          /*c_mod=*/(short)0, acc[t], /*reuse_a=*/false, /*reuse_b=*/false);
    }
  }
#pragma unroll
  for (int t = 0; t < 8; ++t) {
    int col = t * 16 + lm;
#pragma unroll
    for (int v = 0; v < 8; ++v) {
      int row = r0 + v + 8 * kh;                   // C/D layout: M = v + 8*kh
      if (row < nrows) C[(size_t)row * DH + col] = acc[t][v];
    }
  }
}

// ---------------- AGG = XW * dinv^2 + bias (self-loop term) -------------
__global__ __launch_bounds__(256) void k_self_init(const float* __restrict__ XW,
                                                   const float* __restrict__ dinv,
                                                   const float* __restrict__ bias,
                                                   float* __restrict__ AGG, int nrows) {
  int i4 = blockIdx.x * 256 + threadIdx.x;         // over nrows*32 float4s
  if (i4 >= nrows * 32) return;
  int n = i4 >> 5;
  int f0 = (i4 & 31) * 4;
  float d = dinv[n];
  float s = d * d;
  float4 v = ((const float4*)XW)[i4];
  float4 b = *(const float4*)(bias + f0);
  float4 o;
  o.x = v.x * s + b.x;
  o.y = v.y * s + b.y;
  o.z = v.z * s + b.z;
  o.w = v.w * s + b.w;
  ((float4*)AGG)[i4] = o;
}

// ---------------- edge scatter: AGG[dst] += coef * XW[src] --------------
// one thread per (edge, 4-feature chunk): float4 gather + 4 f32 atomics
__global__ __launch_bounds__(256) void k_edge_agg(const int* __restrict__ src,
                                                  const int* __restrict__ dst,
                                                  const float* __restrict__ dinv,
                                                  const float* __restrict__ XW,
                                                  float* __restrict__ AGG, int e) {
  int t = blockIdx.x * 256 + threadIdx.x;
  int eid = t >> 5;
  if (eid >= e) return;
  int f0 = (t & 31) * 4;
  int s = src[eid], d = dst[eid];
  float c = dinv[s] * dinv[d];
  float4 v = *(const float4*)(XW + (size_t)s * DH + f0);
  float* p = AGG + (size_t)d * DH + f0;
  atomicAdd(p + 0, v.x * c);
  atomicAdd(p + 1, v.y * c);
  atomicAdd(p + 2, v.z * c);
  atomicAdd(p + 3, v.w * c);
}

// ---------------- BatchNorm: stats / finalize / apply -------------------
__global__ __launch_bounds__(256) void k_zero_acc(float* __restrict__ acc) {
  acc[threadIdx.x] = 0.0f;
}

// acc[0..127] = sum per feature, acc[128..255] = sum of squares
__global__ __launch_bounds__(256) void k_bn_stats(const float* __restrict__ H,
                                                  float* __restrict__ acc, int nrows) {
  __shared__ float ls[256];
  ls[threadIdx.x] = 0.0f;
  __syncthreads();
  int total4 = nrows * 32;
  int stride = gridDim.x * 256;
  int f0 = (threadIdx.x & 31) * 4;                 // lane -> fixed 4 features
  float s0 = 0, s1 = 0, s2 = 0, s3 = 0;
  float q0 = 0, q1 = 0, q2 = 0, q3 = 0;
  for (int i = blockIdx.x * 256 + threadIdx.x; i < total4; i += stride) {
    float4 v = ((const float4*)H)[i];
    s0 += v.x; q0 += v.x * v.x;
    s1 += v.y; q1 += v.y * v.y;
    s2 += v.z; q2 += v.z * v.z;
    s3 += v.w; q3 += v.w * v.w;
  }
  atomicAdd(&ls[f0 + 0], s0);
  atomicAdd(&ls[f0 + 1], s1);
  atomicAdd(&ls[f0 + 2], s2);
  atomicAdd(&ls[f0 + 3], s3);
  atomicAdd(&ls[128 + f0 + 0], q0);
  atomicAdd(&ls[128 + f0 + 1], q1);
  atomicAdd(&ls[128 + f0 + 2], q2);
  atomicAdd(&ls[128 + f0 + 3], q3);
  __syncthreads();
  atomicAdd(&acc[threadIdx.x], ls[threadIdx.x]);
}

// params[0..127] = scale, params[128..255] = shift
__global__ __launch_bounds__(128) void k_bn_finalize(const float* __restrict__ acc,
                                                     const float* __restrict__ g,
                                                     const float* __restrict__ be,
                                                     float* __restrict__ params, int n) {
  int f = threadIdx.x;
  float invn = 1.0f / (float)n;
  float mean = acc[f] * invn;
  float var = acc[128 + f] * invn - mean * mean;
  float scale = g[f] * rsqrtf(var + BN_EPS);
  params[f] = scale;
  params[128 + f] = be[f] - mean * scale;
}

__global__ __launch_bounds__(256) void k_bn_apply(float* __restrict__ H,
                                                  const float* __restrict__ params,
                                                  int nrows, int relu) {
  int i4 = blockIdx.x * 256 + threadIdx.x;
  if (i4 >= nrows * 32) return;
  int f0 = (i4 & 31) * 4;
  float4 v = ((float4*)H)[i4];
  float4 sc = *(const float4*)(params + f0);
  float4 sh = *(const float4*)(params + 128 + f0);
  v.x = v.x * sc.x + sh.x;
  v.y = v.y * sc.y + sh.y;
  v.z = v.z * sc.z + sh.z;
  v.w = v.w * sc.w + sh.w;
  if (relu) {
    v.x = fmaxf(v.x, 0.0f);
    v.y = fmaxf(v.y, 0.0f);
    v.z = fmaxf(v.z, 0.0f);
    v.w = fmaxf(v.w, 0.0f);
  }
  ((float4*)H)[i4] = v;
}

// ---------------- head: out = log_softmax(H @ Wm + bm) ------------------
// one wave per row, 2 output classes per lane (lanes 0..19 valid)
__global__ __launch_bounds__(256) void k_head(const float* __restrict__ H,
                                              const float* __restrict__ Wm,
                                              const float* __restrict__ bm,
                                              float* __restrict__ out, int nrows) {
  int g = blockIdx.x * 256 + threadIdx.x;
  int row = g >> 5;
  int lane = threadIdx.x & 31;
  if (row >= nrows) return;
  const float4* h4 = (const float4*)(H + (size_t)row * DH);
  int c0 = lane * 2;
  bool valid = (c0 < DOUT);
  int cc = valid ? c0 : 0;
  float a0 = 0.0f, a1 = 0.0f;
  for (int k4 = 0; k4 < 32; ++k4) {
    float4 hv = h4[k4];
    int k = k4 * 4;
    a0 += hv.x * Wm[(k + 0) * DOUT + cc];
    a1 += hv.x * Wm[(k + 0) * DOUT + cc + 1];
    a0 += hv.y * Wm[(k + 1) * DOUT + cc];
    a1 += hv.y * Wm[(k + 1) * DOUT + cc + 1];
    a0 += hv.z * Wm[(k + 2) * DOUT + cc];
    a1 += hv.z * Wm[(k + 2) * DOUT + cc + 1];
    a0 += hv.w * Wm[(k + 3) * DOUT + cc];
    a1 += hv.w * Wm[(k + 3) * DOUT + cc + 1];
  }
  a0 += bm[cc];
  a1 += bm[cc + 1];
  float mx = valid ? fmaxf(a0, a1) : -INFINITY;
  for (int o = 16; o > 0; o >>= 1) mx = fmaxf(mx, __shfl_xor(mx, o, 32));
  float e = valid ? (expf(a0 - mx) + expf(a1 - mx)) : 0.0f;
  for (int o = 16; o > 0; o >>= 1) e += __shfl_xor(e, o, 32);
  float lse = logf(e) + mx;
  if (valid) {
    out[(size_t)row * DOUT + c0] = a0 - lse;
    out[(size_t)row * DOUT + c0 + 1] = a1 - lse;
  }
}

// ---------------- launcher ----------------------------------------------
extern "C" void kernel_launch(void* const* d_in, const int* in_sizes, int n_in,
                              void* d_out, int out_size, void* d_ws, size_t ws_size,
                              hipStream_t stream) {
  const float* x  = (const float*)d_in[0];
  const int* edge = (const int*)d_in[1];
  const float* W1 = (const float*)d_in[2];
  const float* b1 = (const float*)d_in[3];
  const float* W2 = (const float*)d_in[4];
  const float* b2 = (const float*)d_in[5];
  const float* g1 = (const float*)d_in[6];
  const float* be1 = (const float*)d_in[7];
  const float* g2 = (const float*)d_in[8];
  const float* be2 = (const float*)d_in[9];
  const float* g3 = (const float*)d_in[10];
  const float* be3 = (const float*)d_in[11];
  const float* Wm = (const float*)d_in[12];
  const float* bm = (const float*)d_in[13];
  float* out = (float*)d_out;

  const int N = in_sizes[0] / DH;
  const int E = in_sizes[1] / 2;
  const int* src = edge;
  const int* dst = edge + E;

  // workspace layout (floats)
  float* ws = (float*)d_ws;
  const size_t nd = (size_t)N * DH;
  float* B0 = ws;                 // XW buffer
  float* B1 = B0 + nd;            // AGG / hidden buffer
  float* dinv = B1 + nd;          // N floats (deg -> dinv in place)
  float* Wt = dinv + N;           // 128*128 transposed weights
  float* acc = Wt + DH * DH;      // 256 floats (BN sums)
  float* params = acc + 256;      // 256 floats (BN scale/shift)

  const int bN = (N + 255) / 256;
  const int bE = (E + 255) / 256;
  const int bE32 = (E * 32 + 255) / 256;      // 100000 blocks
  const int bElem = (N * 32 + 255) / 256;     // float4 elementwise
  const int bGemm = (N + 127) / 128;
  const int bHead = (N * 32 + 255) / 256;
  const int bStats = 512;

  // degrees with self loops -> dinv
  k_init_deg<<<bN, 256, 0, stream>>>(dinv, N);
  k_deg_accum<<<bE, 256, 0, stream>>>(dst, dinv, E);
  k_dinv<<<bN, 256, 0, stream>>>(dinv, N);

  // ---- layer 1 ----
  k_transpose128<<<DH, DH, 0, stream>>>(W1, Wt);
  k_gemm128_wmma<<<bGemm, 256, 0, stream>>>(x, Wt, B0, N);
  k_self_init<<<bElem, 256, 0, stream>>>(B0, dinv, b1, B1, N);
  k_edge_agg<<<bE32, 256, 0, stream>>>(src, dst, dinv, B0, B1, E);
  k_zero_acc<<<1, 256, 0, stream>>>(acc);
  k_bn_stats<<<bStats, 256, 0, stream>>>(B1, acc, N);
  k_bn_finalize<<<1, 128, 0, stream>>>(acc, g1, be1, params, N);
  k_bn_apply<<<bElem, 256, 0, stream>>>(B1, params, N, 1);

  // ---- layer 2 ----
  k_transpose128<<<DH, DH, 0, stream>>>(W2, Wt);
  k_gemm128_wmma<<<bGemm, 256, 0, stream>>>(B1, Wt, B0, N);
  k_self_init<<<bElem, 256, 0, stream>>>(B0, dinv, b2, B1, N);
  k_edge_agg<<<bE32, 256, 0, stream>>>(src, dst, dinv, B0, B1, E);
  k_zero_acc<<<1, 256, 0, stream>>>(acc);
  k_bn_stats<<<bStats, 256, 0, stream>>>(B1, acc, N);
  k_bn_finalize<<<1, 128, 0, stream>>>(acc, g2, be2, params, N);
  k_bn_apply<<<bElem, 256, 0, stream>>>(B1, params, N, 1);

  // ---- BN3 (no relu) ----
  k_zero_acc<<<1, 256, 0, stream>>>(acc);
  k_bn_stats<<<bStats, 256, 0, stream>>>(B1, acc, N);
  k_bn_finalize<<<1, 128, 0, stream>>>(acc, g3, be3, params, N);
  k_bn_apply<<<bElem, 256, 0, stream>>>(B1, params, N, 0);

  // ---- head + log_softmax ----
  k_head<<<bHead, 256, 0, stream>>>(B1, Wm, bm, out, N);
}